// Loss_15152644620427
// MI455X (gfx1250) — compile-verified
//
#include <hip/hip_runtime.h>
#include <hip/hip_bf16.h>
#include <math.h>

typedef float v2f __attribute__((ext_vector_type(2)));
typedef float v8f __attribute__((ext_vector_type(8)));

#define B_  2
#define N_  50
#define n_  15
#define C_  17
#define HW_ 4096

// workspace layout (float units)
#define WS_MASK 0      // 510 floats: keep[b][j][c]
#define WS_NKP  512    // 30 floats:  num_kp[b][j]
#define WS_ACC  576    // 2048 floats: acc[b][mtile][lane*8+r]

#define LOG2E_F 1.4426950408889634f
#define LN2_F   0.6931471805599453f

// raw hardware TRANS ops, no libm range fixups
__device__ __forceinline__ float fast_rcp(float x) { return __builtin_amdgcn_rcpf(x); }
__device__ __forceinline__ float fast_exp(float x) {           // safe: we only call with x <= 0
    return __builtin_amdgcn_exp2f(x * LOG2E_F);                // v_exp_f32
}
__device__ __forceinline__ float fast_log1pe(float e) {        // log(1+e), e in (0,1]
    return __builtin_amdgcn_logf(1.f + e) * LN2_F;             // v_log_f32; arg in (1,2]
}

__device__ __forceinline__ float softplus_f(float x) {
    float e = fast_exp(-fabsf(x));
    return fmaxf(x, 0.f) + fast_log1pe(e);
}
__device__ __forceinline__ float sigmoid_f(float x) {
    float e   = fast_exp(-fabsf(x));
    float inv = fast_rcp(1.f + e);
    return (x >= 0.f) ? inv : e * inv;
}

// pred-side features: A1 = sp(x)*p^2, A2 = x*p^2, A3 = sp(-x)*(1-p)^2
__device__ __forceinline__ void pred_feats(float x, float& a1, float& a2, float& a3) {
    float e      = fast_exp(-fabsf(x));        // v_exp_f32 (TRANS)
    float inv    = fast_rcp(1.f + e);          // v_rcp_f32 (TRANS)
    float sp     = fmaxf(x, 0.f) + fast_log1pe(e);   // softplus(x)
    float p      = (x >= 0.f) ? inv : e * inv; // sigmoid(x)
    float q      = (x >= 0.f) ? e * inv : inv; // 1 - sigmoid(x)
    float p2 = p * p;
    a1 = sp * p2;
    a2 = x * p2;
    a3 = (sp - x) * q * q;                     // softplus(-x)*(1-p)^2
}

// gt-side features with mask folded in; G2 carries the minus sign.
// keep==0 (pad column or masked channel) forces all three to zero.
__device__ __forceinline__ void gt_feats(float t, float keep, float& g1, float& g2, float& g3) {
    float omt = 1.f - t;
    float o2  = omt * omt;
    float r4  = o2 * o2;                       // (1-t)^4
    bool  eq  = (t == 1.f);
    g1 = eq ? 0.f : keep * r4;
    g2 = eq ? 0.f : -(keep * t * r4);
    g3 = eq ? keep : 0.f;
}

// 1) keep-flag per (b, j, c): sum gh over HW, keep = (sum != 0)
__global__ __launch_bounds__(256) void mask_kernel(const float* __restrict__ gh,
                                                   float* __restrict__ ws) {
    int idx = blockIdx.x;                      // (b*n_ + j)*C_ + c, 0..509
    const float* src = gh + (size_t)idx * HW_;
    float s = 0.f;
    for (int t = threadIdx.x; t < HW_; t += 256) s += src[t];
    __shared__ float red[256];
    red[threadIdx.x] = s;
    __syncthreads();
    for (int off = 128; off > 0; off >>= 1) {
        if (threadIdx.x < off) red[threadIdx.x] += red[threadIdx.x + off];
        __syncthreads();
    }
    if (threadIdx.x == 0) ws[WS_MASK + idx] = (red[0] != 0.f) ? 1.f : 0.f;
}

// 2) zero the WMMA accumulator region (ws is poisoned by the harness)
__global__ __launch_bounds__(256) void zero_kernel(float* __restrict__ ws) {
    int i = blockIdx.x * 256 + threadIdx.x;
    if (i < 2048) ws[WS_ACC + i] = 0.f;
}

// 3) num_kp per (b, j) = max(1, #keep)
__global__ void nkp_kernel(float* __restrict__ ws) {
    int t = threadIdx.x;
    if (t < B_ * n_) {
        float cnt = 0.f;
        for (int c = 0; c < C_; ++c) cnt += ws[WS_MASK + t * C_ + c];
        ws[WS_NKP + t] = fmaxf(cnt, 1.f);
    }
}

// 4) fused feature + WMMA GEMM over k = (c, hw)
//    grid: x = 68 (17 c-planes * 4 chunks of 1024 cols), y = 4 M-tiles, z = 2 batches
__global__ __launch_bounds__(256) void hm_gemm_kernel(const float* __restrict__ ph,
                                                      const float* __restrict__ gh,
                                                      float* __restrict__ ws) {
    const int b    = blockIdx.z;
    const int mt   = blockIdx.y;
    const int c    = blockIdx.x >> 2;
    const int kblk = (blockIdx.x & 3) * 1024;
    const int w    = threadIdx.x >> 5;
    const int lane = threadIdx.x & 31;
    const int row  = lane & 15;                // A: M index / B: N index
    const int hi   = lane >> 4;                // 0 -> K{0,1}, 1 -> K{2,3}

    const int  ig   = mt * 16 + row;           // pred row
    const bool a_ok = (ig < N_);
    const bool g_ok = (row < n_);
    // keep==0 for pad column j==15 zeroes the whole B column; pad A rows
    // (ig>=50) produce garbage C rows that finalize never reads, so both
    // loads below can be unconditional from clamped (valid) rows.
    const float keep = g_ok ? ws[WS_MASK + (b * n_ + row) * C_ + c] : 0.f;

    const float* aRow = ph + (((size_t)(b * N_ + (a_ok ? ig : 0)) * C_ + c) << 12);
    const float* gRow = gh + (((size_t)(b * n_ + (g_ok ? row : 0)) * C_ + c) << 12);

    int k0 = kblk + w * 128 + hi * 2;          // each wave covers 128 columns
    v8f acc = {0.f, 0.f, 0.f, 0.f, 0.f, 0.f, 0.f, 0.f};

#pragma unroll 4
    for (int it = 0; it < 32; ++it, k0 += 4) {
        float2 xv = *(const float2*)(aRow + k0);   // global_load_b64, no EXEC branch
        float2 tv = *(const float2*)(gRow + k0);

        v2f A1, A2, A3, G1, G2, G3;
        {
            float a1, a2, a3;
            pred_feats(xv.x, a1, a2, a3); A1[0] = a1; A2[0] = a2; A3[0] = a3;
            pred_feats(xv.y, a1, a2, a3); A1[1] = a1; A2[1] = a2; A3[1] = a3;
        }
        {
            float g1, g2, g3;
            gt_feats(tv.x, keep, g1, g2, g3); G1[0] = g1; G2[0] = g2; G3[0] = g3;
            gt_feats(tv.y, keep, g1, g2, g3); G1[1] = g1; G2[1] = g2; G3[1] = g3;
        }

        acc = __builtin_amdgcn_wmma_f32_16x16x4_f32(false, A1, false, G1,
                                                    (short)0, acc, false, false);
        acc = __builtin_amdgcn_wmma_f32_16x16x4_f32(false, A2, false, G2,
                                                    (short)0, acc, false, false);
        acc = __builtin_amdgcn_wmma_f32_16x16x4_f32(false, A3, false, G3,
                                                    (short)0, acc, false, false);
    }

    // cross-wave reduce (8 waves) then atomic-add into ws accumulator
    __shared__ float red[8 * 256];
#pragma unroll
    for (int r = 0; r < 8; ++r) red[w * 256 + lane * 8 + r] = acc[r];
    __syncthreads();
    float s = 0.f;
    const int t = threadIdx.x;
#pragma unroll
    for (int ww = 0; ww < 8; ++ww) s += red[ww * 256 + t];
    atomicAdd(&ws[WS_ACC + ((b * 4 + mt) << 8) + t], s);
}

// 5) finalize: out[b,i,j] = 2*hm/num_kp + score(b,i) + off(b,i,j)
__global__ __launch_bounds__(256) void finalize_kernel(const float* __restrict__ ps,
                                                       const float* __restrict__ po,
                                                       const float* __restrict__ go,
                                                       const float* __restrict__ ws,
                                                       float* __restrict__ out) {
    int idx = blockIdx.x * 256 + threadIdx.x;
    if (idx >= B_ * N_ * n_) return;
    int j = idx % n_;
    int i = (idx / n_) % N_;
    int b = idx / (n_ * N_);

    float nkp  = ws[WS_NKP + b * n_ + j];
    float rnkp = fast_rcp(nkp);

    // decode WMMA C layout: M=row -> VGPR r = row&7, lane = (row>>3)*16 + N
    int mt = i >> 4, row = i & 15;
    int lane = ((row >> 3) << 4) | j;
    int flat = lane * 8 + (row & 7);
    float hm = ws[WS_ACC + ((b * 4 + mt) << 8) + flat] * rnkp;

    // score cost: ALPHA * softplus(-s) * (1 - sigmoid(s))^2
    float s  = ps[b * N_ + i];
    float q  = 1.f - sigmoid_f(s);
    float sc = 0.25f * softplus_f(-s) * q * q;

    // offset cost
    float oc = 0.f;
    for (int c = 0; c < C_; ++c) {
        float keep = ws[WS_MASK + (b * n_ + j) * C_ + c];
        const float* pp = po + ((size_t)(b * N_ + i) * C_ + c) * 2;
        const float* gg = go + ((size_t)(b * n_ + j) * C_ + c) * 2;
        float d0 = sigmoid_f(pp[0]) - gg[0];
        float d1 = sigmoid_f(pp[1]) - gg[1];
        oc += keep * (d0 * d0 + d1 * d1);
    }
    oc = oc * rnkp * 0.5f;

    out[idx] = 2.f * hm + sc + oc;
}

extern "C" void kernel_launch(void* const* d_in, const int* in_sizes, int n_in,
                              void* d_out, int out_size, void* d_ws, size_t ws_size,
                              hipStream_t stream) {
    const float* ph = (const float*)d_in[0];  // [2,50,17,64,64]
    const float* ps = (const float*)d_in[1];  // [2,50,1]
    const float* po = (const float*)d_in[2];  // [2,50,17,2]
    const float* gh = (const float*)d_in[3];  // [2,15,17,64,64]
    const float* go = (const float*)d_in[4];  // [2,15,17,2]
    float* out = (float*)d_out;               // [2,50,15]
    float* ws  = (float*)d_ws;

    mask_kernel<<<B_ * n_ * C_, 256, 0, stream>>>(gh, ws);
    zero_kernel<<<8, 256, 0, stream>>>(ws);
    nkp_kernel<<<1, 32, 0, stream>>>(ws);
    hm_gemm_kernel<<<dim3(68, 4, 2), 256, 0, stream>>>(ph, gh, ws);
    finalize_kernel<<<6, 256, 0, stream>>>(ps, po, go, ws, out);
}